// WordDropout_16363825398135
// MI455X (gfx1250) — compile-verified
//
#include <hip/hip_runtime.h>

// Masked-mean embedding pooling for MI455X (gfx1250).
// B=16384 rows, L=200 tokens/row, D=20 dims, V=1e6 vocab.
// Memory-bound gather/reduce: ~275 MB traffic, 80 MB table is L2-resident.

constexpr int Ln  = 200;
constexpr int Dn  = 20;
constexpr int WPB = 8;          // waves (rows) per block
constexpr int BLK = WPB * 32;   // 256 threads = 8 wave32

__device__ __forceinline__ float wave_sum(float v) {
  v += __shfl_xor(v, 16, 32);
  v += __shfl_xor(v, 8, 32);
  v += __shfl_xor(v, 4, 32);
  v += __shfl_xor(v, 2, 32);
  v += __shfl_xor(v, 1, 32);
  return v;
}

__global__ __launch_bounds__(BLK)
void masked_mean_pool(const int* __restrict__ ids,
                      const float* __restrict__ emb,
                      float* __restrict__ out,
                      int nrows) {
  __shared__ int sIdx[WPB * Ln];  // 6400 B

  const int tid  = threadIdx.x;
  const int row0 = blockIdx.x * WPB;

  // ---- async stage this block's indices into LDS (ASYNCcnt path) ----
  {
    const int* gIdx = ids + (size_t)row0 * Ln;
    const int lim = min(WPB * Ln, (nrows - row0) * Ln);
    for (int i = tid; i < lim; i += BLK) {
      __builtin_amdgcn_global_load_async_to_lds_b32(
          (__attribute__((address_space(1))) int*)(gIdx + i),
          (__attribute__((address_space(3))) int*)(&sIdx[i]),
          0, 0);
    }
  }
  __builtin_amdgcn_s_wait_asynccnt(0);
  __syncthreads();

  const int wave = tid >> 5;
  const int lane = tid & 31;
  const int b    = row0 + wave;
  if (b >= nrows) return;
  const int* widx = &sIdx[wave * Ln];

  // ---- prefetch pass: issue gather prefetches for this lane's rows ----
  for (int l = lane; l < Ln; l += 32) {
    int v = widx[l];
    if (v != 0) __builtin_prefetch(emb + (size_t)v * Dn, 0, 3);
  }

  // ---- gather + masked accumulate ----
  float a[Dn];
  #pragma unroll
  for (int d = 0; d < Dn; ++d) a[d] = 0.0f;
  float cnt = 0.0f;

  #pragma unroll 2
  for (int l = lane; l < Ln; l += 32) {
    const int v = widx[l];
    if (v == 0) continue;  // padding row is exactly zero -> filtered anyway
    const float4* r4 = (const float4*)(emb + (size_t)v * Dn);  // 16B-aligned (80B rows)
    const float4 r0 = r4[0], r1 = r4[1], r2 = r4[2], r3 = r4[3], rv = r4[4];
    const float s = (r0.x + r0.y + r0.z + r0.w)
                  + (r1.x + r1.y + r1.z + r1.w)
                  + (r2.x + r2.y + r2.z + r2.w)
                  + (r3.x + r3.y + r3.z + r3.w)
                  + (rv.x + rv.y + rv.z + rv.w);
    if (s != 0.0f) {
      cnt += 1.0f;
      a[0]  += r0.x; a[1]  += r0.y; a[2]  += r0.z; a[3]  += r0.w;
      a[4]  += r1.x; a[5]  += r1.y; a[6]  += r1.z; a[7]  += r1.w;
      a[8]  += r2.x; a[9]  += r2.y; a[10] += r2.z; a[11] += r2.w;
      a[12] += r3.x; a[13] += r3.y; a[14] += r3.z; a[15] += r3.w;
      a[16] += rv.x; a[17] += rv.y; a[18] += rv.z; a[19] += rv.w;
    }
  }

  // ---- wave32 butterfly reduction (21 values) ----
  #pragma unroll
  for (int d = 0; d < Dn; ++d) a[d] = wave_sum(a[d]);
  cnt = wave_sum(cnt);

  if (lane == 0) {
    const float inv = 1.0f / fmaxf(cnt, 1.0f);   // cnt==0 -> a[]==0 -> zeros out
    float4* o = (float4*)(out + (size_t)b * Dn); // 80B rows, 16B-aligned
    o[0] = make_float4(a[0]  * inv, a[1]  * inv, a[2]  * inv, a[3]  * inv);
    o[1] = make_float4(a[4]  * inv, a[5]  * inv, a[6]  * inv, a[7]  * inv);
    o[2] = make_float4(a[8]  * inv, a[9]  * inv, a[10] * inv, a[11] * inv);
    o[3] = make_float4(a[12] * inv, a[13] * inv, a[14] * inv, a[15] * inv);
    o[4] = make_float4(a[16] * inv, a[17] * inv, a[18] * inv, a[19] * inv);
  }
}

extern "C" void kernel_launch(void* const* d_in, const int* in_sizes, int n_in,
                              void* d_out, int out_size, void* d_ws, size_t ws_size,
                              hipStream_t stream) {
  const int*   ids = (const int*)d_in[0];    // [B, L] int32
  const float* emb = (const float*)d_in[1];  // [V, D] f32
  float*       out = (float*)d_out;          // [B, D] f32

  const int nrows = in_sizes[0] / Ln;        // B
  const int grid  = (nrows + WPB - 1) / WPB;
  masked_mean_pool<<<grid, BLK, 0, stream>>>(ids, emb, out, nrows);
}